// GCNN_9577777070402
// MI455X (gfx1250) — compile-verified
//
#include <hip/hip_runtime.h>

// ---------------------------------------------------------------------------
// GraphConv GNN for MI455X (gfx1250, wave32).
// Strategy:
//   xrel = x @ c1_rel_w            (WMMA f32 16x16x4, K=64)
//   agg1 = scatter_add(xrel[src]*ew -> dst)   (coalesced f32 atomics, L2)
//   h1   = relu(agg1 + b1 + x @ c1_root_w)    (WMMA, C-epilogue fused)
//   h1r  = h1 @ c2_rel_w           (WMMA, K=32)
//   agg2 = scatter_add(h1r[src]*ew -> dst)
//   h2   = relu(agg2 + b2 + h1 @ c2_root_w)   (WMMA, epilogue pools into
//                                              per-graph sums via atomics)
//   out  = tiny MLP over 256 graphs (one block)
// ---------------------------------------------------------------------------

typedef float v2f __attribute__((ext_vector_type(2)));
typedef float v8f __attribute__((ext_vector_type(8)));

#define FW 32  // hidden width of every node-feature buffer

// ---------------------------------------------------------------------------
// Zero fill
// ---------------------------------------------------------------------------
__global__ void zero_kernel(float* __restrict__ p, long n) {
    long i = (long)blockIdx.x * blockDim.x + threadIdx.x;
    long stride = (long)gridDim.x * blockDim.x;
    for (; i < n; i += stride) p[i] = 0.0f;
}

// ---------------------------------------------------------------------------
// WMMA GEMM: out[N,32] = A[N,KDIM] @ W[KDIM,32]  (+ optional epilogue)
// MODE 0: store raw
// MODE 1: out = relu(acc + Cin + bias), store
// MODE 2: r = relu(acc + Cin + bias); atomically pool r into pool[batch[row]]
// One wave handles one 16-row tile; two 16x16 accumulators cover 32 cols.
// f32 WMMA layouts (ISA 7.12.2):
//   A 16x4 : lane l (m=l&15, hi=l>>4) holds A[m][k+2hi], A[m][k+2hi+1]
//   B 4x16 : lane l holds W[k+2hi][n], W[k+2hi+1][n]   (n=l&15)
//   C/D    : vgpr v, lane l -> row (v + 8*hi), col n
// Requires N % 16 == 0 (true here: 100000 = 6250*16).
// ---------------------------------------------------------------------------
template<int KDIM, int MODE>
__global__ __launch_bounds__(128)
void gemm_wmma(const float* __restrict__ A,
               const float* __restrict__ W,
               const float* __restrict__ Cin,
               const float* __restrict__ bias,
               const long long* __restrict__ batch,
               float* __restrict__ out,
               float* __restrict__ pool,
               float* __restrict__ cnt,
               int ntiles) {
    __shared__ float sW[KDIM * FW];
    for (int i = threadIdx.x; i < KDIM * FW; i += blockDim.x) sW[i] = W[i];
    __syncthreads();

    const int wave = threadIdx.x >> 5;
    const int lane = threadIdx.x & 31;
    const int tile = blockIdx.x * 4 + wave;
    if (tile >= ntiles) return;            // wave-uniform exit: EXEC stays full

    const int row0 = tile * 16;
    const int m  = lane & 15;
    const int hi = lane >> 4;
    const int n  = m;

    v8f acc0 = {};
    v8f acc1 = {};
    const float* arow = A + (size_t)(row0 + m) * KDIM;

    #pragma unroll
    for (int k = 0; k < KDIM; k += 4) {
        const int ka = k + 2 * hi;
        v2f a;  a.x  = arow[ka];              a.y  = arow[ka + 1];
        v2f b0; b0.x = sW[ka * FW + n];       b0.y = sW[(ka + 1) * FW + n];
        v2f b1; b1.x = sW[ka * FW + 16 + n];  b1.y = sW[(ka + 1) * FW + 16 + n];
        acc0 = __builtin_amdgcn_wmma_f32_16x16x4_f32(false, a, false, b0,
                                                     (short)0, acc0, false, false);
        acc1 = __builtin_amdgcn_wmma_f32_16x16x4_f32(false, a, false, b1,
                                                     (short)0, acc1, false, false);
    }

    if (MODE == 0) {
        #pragma unroll
        for (int v = 0; v < 8; ++v) {
            const int row = row0 + v + 8 * hi;
            out[(size_t)row * FW + n]      = acc0[v];
            out[(size_t)row * FW + 16 + n] = acc1[v];
        }
    } else {
        const float bn0 = bias[n];
        const float bn1 = bias[16 + n];
        #pragma unroll
        for (int v = 0; v < 8; ++v) {
            const int row = row0 + v + 8 * hi;
            float r0 = acc0[v] + Cin[(size_t)row * FW + n]      + bn0;
            float r1 = acc1[v] + Cin[(size_t)row * FW + 16 + n] + bn1;
            r0 = fmaxf(r0, 0.0f);
            r1 = fmaxf(r1, 0.0f);
            if (MODE == 1) {
                out[(size_t)row * FW + n]      = r0;
                out[(size_t)row * FW + 16 + n] = r1;
            } else {
                const int g = (int)batch[row];
                unsafeAtomicAdd(&pool[g * FW + n],      r0);
                unsafeAtomicAdd(&pool[g * FW + 16 + n], r1);
                if (n == 0) unsafeAtomicAdd(&cnt[g], 1.0f);
            }
        }
    }
}

// ---------------------------------------------------------------------------
// Edge scatter: agg[dst,f] += xin[src,f] * ew.  32 lanes = one edge, so the
// src-gather and dst-atomic are each one coalesced 128B transaction per wave.
// ---------------------------------------------------------------------------
__global__ __launch_bounds__(256)
void scatter_kernel(const float* __restrict__ xin,
                    const long long* __restrict__ ei,   // [2,E]
                    const float* __restrict__ ew,
                    float* __restrict__ agg,
                    int E) {
    const long t = (long)blockIdx.x * blockDim.x + threadIdx.x;
    const int  e = (int)(t >> 5);
    const int  f = (int)(t & 31);
    if (e >= E) return;
    const int s = (int)ei[e];
    const int d = (int)ei[(size_t)E + e];
    const float w = ew[e];
    const float v = xin[(size_t)s * FW + f] * w;
    unsafeAtomicAdd(&agg[(size_t)d * FW + f], v);
}

// ---------------------------------------------------------------------------
// Final MLP over G=256 graphs. One thread per graph.
// ---------------------------------------------------------------------------
__global__ __launch_bounds__(256)
void mlp_kernel(const float* __restrict__ pool, const float* __restrict__ cnt,
                const float* __restrict__ W1, const float* __restrict__ b1,
                const float* __restrict__ W2, const float* __restrict__ b2,
                const float* __restrict__ Wf, const float* __restrict__ bf,
                float* __restrict__ outp, int G) {
    __shared__ float sW1[32 * 32], sW2[32 * 16], sWf[16];
    for (int i = threadIdx.x; i < 32 * 32; i += blockDim.x) sW1[i] = W1[i];
    for (int i = threadIdx.x; i < 32 * 16; i += blockDim.x) sW2[i] = W2[i];
    if (threadIdx.x < 16) sWf[threadIdx.x] = Wf[threadIdx.x];
    __syncthreads();

    const int g = blockIdx.x * blockDim.x + threadIdx.x;
    if (g >= G) return;

    const float inv = 1.0f / fmaxf(cnt[g], 1.0f);
    float a[32];
    #pragma unroll
    for (int i = 0; i < 32; ++i) a[i] = pool[g * 32 + i] * inv;

    float t1[32];
    #pragma unroll
    for (int j = 0; j < 32; ++j) {
        float s = b1[j];
        #pragma unroll
        for (int i = 0; i < 32; ++i) s = fmaf(a[i], sW1[i * 32 + j], s);
        t1[j] = fmaxf(s, 0.0f);
    }
    float o = bf[0];
    #pragma unroll
    for (int j = 0; j < 16; ++j) {
        float s = b2[j];
        #pragma unroll
        for (int i = 0; i < 32; ++i) s = fmaf(t1[i], sW2[i * 16 + j], s);
        o = fmaf(fmaxf(s, 0.0f), sWf[j], o);
    }
    outp[g] = o;
}

// ---------------------------------------------------------------------------
// Launch
// ---------------------------------------------------------------------------
extern "C" void kernel_launch(void* const* d_in, const int* in_sizes, int n_in,
                              void* d_out, int out_size, void* d_ws, size_t ws_size,
                              hipStream_t stream) {
    const float*     x       = (const float*)d_in[0];
    const long long* ei      = (const long long*)d_in[1];
    const float*     ew      = (const float*)d_in[2];
    const long long* batch   = (const long long*)d_in[3];
    const float*     c1_rw   = (const float*)d_in[4];
    const float*     c1_rb   = (const float*)d_in[5];
    const float*     c1_ow   = (const float*)d_in[6];
    const float*     c2_rw   = (const float*)d_in[7];
    const float*     c2_rb   = (const float*)d_in[8];
    const float*     c2_ow   = (const float*)d_in[9];
    const float*     l1w     = (const float*)d_in[10];
    const float*     l1b     = (const float*)d_in[11];
    const float*     l2w     = (const float*)d_in[12];
    const float*     l2b     = (const float*)d_in[13];
    const float*     lw      = (const float*)d_in[14];
    const float*     lb      = (const float*)d_in[15];
    float*           outp    = (float*)d_out;

    const int N = in_sizes[0] / 64;     // 100000
    const int E = in_sizes[2];          // 1600000
    const int G = out_size;             // 256

    float* ws     = (float*)d_ws;
    float* bufA   = ws;                     // [N,32] xrel, reused as h1rel
    float* bufAgg = ws + (size_t)N * FW;    // [N,32] agg1, reused as agg2
    float* bufH1  = ws + 2 * (size_t)N * FW;// [N,32]
    float* pool   = ws + 3 * (size_t)N * FW;// [G,32]
    float* cnt    = pool + (size_t)G * FW;  // [G]

    const int ntiles     = N / 16;                      // N % 16 == 0
    const int gemmBlocks = (ntiles + 3) / 4;            // 4 waves/block
    const long scatterThreads = (long)E * 32;
    const int scatterBlocks   = (int)((scatterThreads + 255) / 256);

    // zero agg1 and pool/cnt
    zero_kernel<<<2048, 256, 0, stream>>>(bufAgg, (long)N * FW);
    zero_kernel<<<64, 256, 0, stream>>>(pool, (long)G * FW + G);

    // xrel = x @ c1_rel_w
    gemm_wmma<64, 0><<<gemmBlocks, 128, 0, stream>>>(
        x, c1_rw, nullptr, nullptr, nullptr, bufA, nullptr, nullptr, ntiles);

    // agg1 += xrel[src] * ew  (scatter to dst)
    scatter_kernel<<<scatterBlocks, 256, 0, stream>>>(bufA, ei, ew, bufAgg, E);

    // h1 = relu(agg1 + b1 + x @ c1_root_w)
    gemm_wmma<64, 1><<<gemmBlocks, 128, 0, stream>>>(
        x, c1_ow, bufAgg, c1_rb, nullptr, bufH1, nullptr, nullptr, ntiles);

    // re-zero agg for layer 2
    zero_kernel<<<2048, 256, 0, stream>>>(bufAgg, (long)N * FW);

    // h1rel = h1 @ c2_rel_w
    gemm_wmma<32, 0><<<gemmBlocks, 128, 0, stream>>>(
        bufH1, c2_rw, nullptr, nullptr, nullptr, bufA, nullptr, nullptr, ntiles);

    // agg2 += h1rel[src] * ew
    scatter_kernel<<<scatterBlocks, 256, 0, stream>>>(bufA, ei, ew, bufAgg, E);

    // h2 = relu(agg2 + b2 + h1 @ c2_root_w), pooled per graph
    gemm_wmma<32, 2><<<gemmBlocks, 128, 0, stream>>>(
        bufH1, c2_ow, bufAgg, c2_rb, batch, nullptr, pool, cnt, ntiles);

    // final MLP
    mlp_kernel<<<(G + 255) / 256, 256, 0, stream>>>(
        pool, cnt, l1w, l1b, l2w, l2b, lw, lb, outp, G);
}